// Decoder_36335423324794
// MI455X (gfx1250) — compile-verified
//
#include <hip/hip_runtime.h>
#include <math.h>

#define VOCAB 50257
#define HDIM  1024
#define MAXLEN 128

// ---- workspace layout (floats) ----
#define WS_LOGITS   0        // 50432 (padded)
#define WS_ATTNLOG  50432    // 128
#define WS_ATTNW    50560    // 128
#define WS_ATTNAP   50688    // 1024
#define WS_G        51712    // 1024
#define WS_GI       52736    // 3072
#define WS_GH       55808    // 3072
#define WS_HNEW     58880    // 1024
#define WS_RED      59904    // 2  (max, log-sum)
#define WS_PART     59908    // 128 (64 x {max, sumexp})
#define LSE_BLOCKS  64

typedef float v2f __attribute__((ext_vector_type(2)));
typedef float v4f __attribute__((ext_vector_type(4)));
typedef float v8f __attribute__((ext_vector_type(8)));

__device__ __forceinline__ float wave_sum(float v) {
    v += __shfl_xor(v, 16, 32);
    v += __shfl_xor(v, 8, 32);
    v += __shfl_xor(v, 4, 32);
    v += __shfl_xor(v, 2, 32);
    v += __shfl_xor(v, 1, 32);
    return v;
}

// K1: attention logits  attnlog[t] = [e,h0] . W_attn[t,:] + b_attn[t]
__global__ void k_attn_logits(const int* __restrict__ x, const float* __restrict__ h,
                              const float* __restrict__ emb, const float* __restrict__ W_attn,
                              const float* __restrict__ b_attn, float* __restrict__ attnlog) {
    __shared__ float red[256];
    const int t = blockIdx.x, tid = threadIdx.x;
    const float* e  = emb + (long)x[0] * HDIM;
    const float* wr = W_attn + (long)t * (2 * HDIM);
    float acc = 0.f;
    for (int j = tid; j < HDIM; j += 256)
        acc += e[j] * wr[j] + h[j] * wr[HDIM + j];
    red[tid] = acc; __syncthreads();
    for (int s = 128; s > 0; s >>= 1) { if (tid < s) red[tid] += red[tid + s]; __syncthreads(); }
    if (tid == 0) attnlog[t] = red[0] + b_attn[t];
}

// K2: softmax over 128 attention logits
__global__ void k_softmax128(const float* __restrict__ attnlog, float* __restrict__ attnw,
                             float* __restrict__ out_attnw) {
    __shared__ float red[128];
    const int t = threadIdx.x;
    const float v = attnlog[t];
    red[t] = v; __syncthreads();
    for (int s = 64; s > 0; s >>= 1) { if (t < s) red[t] = fmaxf(red[t], red[t + s]); __syncthreads(); }
    const float mx = red[0]; __syncthreads();
    const float e = expf(v - mx);
    red[t] = e; __syncthreads();
    for (int s = 64; s > 0; s >>= 1) { if (t < s) red[t] += red[t + s]; __syncthreads(); }
    const float w = e / red[0];
    attnw[t] = w;
    out_attnw[t] = w;
}

// K3: attn_applied[j] = sum_t attnw[t] * enc[t,j]   (grid 8 x 128)
__global__ void k_attn_applied(const float* __restrict__ attnw, const float* __restrict__ enc,
                               float* __restrict__ attnap) {
    __shared__ float w[MAXLEN];
    if (threadIdx.x < MAXLEN) w[threadIdx.x] = attnw[threadIdx.x];
    __syncthreads();
    const int j = blockIdx.x * 128 + threadIdx.x;
    float acc = 0.f;
    for (int t = 0; t < MAXLEN; ++t) acc += w[t] * enc[(long)t * HDIM + j];
    attnap[j] = acc;
}

// K4: g[r] = relu([e, attnap] . W_comb[r,:] + b_comb[r])  -- wave per row
__global__ void k_combine(const int* __restrict__ x, const float* __restrict__ emb,
                          const float* __restrict__ attnap, const float* __restrict__ W_comb,
                          const float* __restrict__ b_comb, float* __restrict__ g) {
    const int wid = blockIdx.x * 8 + (threadIdx.x >> 5);   // 0..1023
    const int lane = threadIdx.x & 31;
    const float* e  = emb + (long)x[0] * HDIM;
    const float* wr = W_comb + (long)wid * (2 * HDIM);
    float acc = 0.f;
    for (int j = lane * 4; j < HDIM; j += 128) {
        float4 a = *(const float4*)(wr + j);
        float4 s = *(const float4*)(e + j);
        acc += a.x * s.x + a.y * s.y + a.z * s.z + a.w * s.w;
    }
    for (int j = lane * 4; j < HDIM; j += 128) {
        float4 a = *(const float4*)(wr + HDIM + j);
        float4 s = *(const float4*)(attnap + j);
        acc += a.x * s.x + a.y * s.y + a.z * s.z + a.w * s.w;
    }
    acc = wave_sum(acc);
    if (lane == 0) g[wid] = fmaxf(acc + b_comb[wid], 0.f);
}

// K5: gi = w_ih @ g + b_ih ; gh = w_hh @ h0 + b_hh  -- wave per row, 6144 waves
__global__ void k_gru_matvec(const float* __restrict__ g, const float* __restrict__ h,
                             const float* __restrict__ w_ih, const float* __restrict__ w_hh,
                             const float* __restrict__ b_ih, const float* __restrict__ b_hh,
                             float* __restrict__ gi, float* __restrict__ gh) {
    const int wid = blockIdx.x * 8 + (threadIdx.x >> 5);   // 0..6143
    const int lane = threadIdx.x & 31;
    const bool is_ih = (wid < 3 * HDIM);
    const int row = is_ih ? wid : wid - 3 * HDIM;
    const float* mat = (is_ih ? w_ih : w_hh) + (long)row * HDIM;
    const float* src = is_ih ? g : h;
    float acc = 0.f;
    for (int j = lane * 4; j < HDIM; j += 128) {
        float4 a = *(const float4*)(mat + j);
        float4 s = *(const float4*)(src + j);
        acc += a.x * s.x + a.y * s.y + a.z * s.z + a.w * s.w;
    }
    acc = wave_sum(acc);
    if (lane == 0) {
        if (is_ih) gi[row] = acc + b_ih[row];
        else       gh[row] = acc + b_hh[row];
    }
}

// K6: GRU gates -> h_new
__global__ void k_gates(const float* __restrict__ gi, const float* __restrict__ gh,
                        const float* __restrict__ h, float* __restrict__ hnew,
                        float* __restrict__ out_h) {
    const int i = blockIdx.x * 256 + threadIdx.x;
    if (i >= HDIM) return;
    const float r  = 1.f / (1.f + expf(-(gi[i] + gh[i])));
    const float z  = 1.f / (1.f + expf(-(gi[HDIM + i] + gh[HDIM + i])));
    const float n  = tanhf(gi[2 * HDIM + i] + r * gh[2 * HDIM + i]);
    const float hn = (1.f - z) * n + z * h[i];
    hnew[i]  = hn;
    out_h[i] = hn;
}

// K7: logits = W_out @ h_new + b_out via V_WMMA_F32_16X16X4_F32.
// Wave -> 16 rows of W_out. A = streamed W tile (non-temporal: 206MB > 192MB L2,
// keep L2 for the reusable small weights), B = h broadcast to all 16 columns.
// WMMA performs the cross-lane K reduction; dual accumulators carry the K loop.
// EXEC kept all-1s (row clamp; only stores masked).
__global__ void k_wout_wmma(const float* __restrict__ W, const float* __restrict__ bias,
                            const float* __restrict__ h, float* __restrict__ logits) {
    __shared__ float hs[HDIM];
    for (int i = threadIdx.x; i < HDIM; i += 256) hs[i] = h[i];
    __syncthreads();
    const int lane = threadIdx.x & 31;
    const int tile = blockIdx.x * 8 + (threadIdx.x >> 5);
    long row = (long)tile * 16 + (lane & 15);
    if (row > VOCAB - 1) row = VOCAB - 1;                 // clamp: uniform EXEC for WMMA
    const int half = (lane >> 4) << 2;                    // 0 or 4
    const v4f* wrow = (const v4f*)(W + row * HDIM + half);
    const float* hp = hs + half;
    v8f acc0 = {0.f, 0.f, 0.f, 0.f, 0.f, 0.f, 0.f, 0.f};
    v8f acc1 = {0.f, 0.f, 0.f, 0.f, 0.f, 0.f, 0.f, 0.f};
#pragma unroll 4
    for (int s = 0; s < HDIM / 8; ++s) {
        const v4f w4 = __builtin_nontemporal_load(wrow + s * 2);   // TH=NT stream
        const v4f g4 = *(const v4f*)(hp + s * 8);
        v2f a1 = {w4.x, w4.y}, b1 = {g4.x, g4.y};
        v2f a2 = {w4.z, w4.w}, b2 = {g4.z, g4.w};
        acc0 = __builtin_amdgcn_wmma_f32_16x16x4_f32(false, a1, false, b1, (short)0, acc0, false, false);
        acc1 = __builtin_amdgcn_wmma_f32_16x16x4_f32(false, a2, false, b2, (short)0, acc1, false, false);
    }
    const v8f acc = acc0 + acc1;
    // D layout: VGPR r holds M=r (lanes 0-15) / M=8+r (lanes 16-31); all N equal.
    if ((lane & 15) == 0) {
        const long v0 = (long)tile * 16 + ((lane >> 4) << 3);
#pragma unroll
        for (int r = 0; r < 8; ++r) {
            const long v = v0 + r;
            if (v < VOCAB) logits[v] = acc[r] + bias[v];
        }
    }
}

// K8a: per-block streaming logsumexp partials over the 50257 logits
__global__ void k_lse_part(const float* __restrict__ logits, float* __restrict__ part) {
    __shared__ float red[256];
    const int tid = threadIdx.x;
    const int base = blockIdx.x * 256 + tid;
    const int stride = LSE_BLOCKS * 256;
    float mx = -3.402823466e38f;
    for (int v = base; v < VOCAB; v += stride) mx = fmaxf(mx, logits[v]);
    red[tid] = mx; __syncthreads();
    for (int s = 128; s > 0; s >>= 1) { if (tid < s) red[tid] = fmaxf(red[tid], red[tid + s]); __syncthreads(); }
    mx = red[0]; __syncthreads();
    float sum = 0.f;
    for (int v = base; v < VOCAB; v += stride) sum += expf(logits[v] - mx);
    red[tid] = sum; __syncthreads();
    for (int s = 128; s > 0; s >>= 1) { if (tid < s) red[tid] += red[tid + s]; __syncthreads(); }
    if (tid == 0) { part[blockIdx.x * 2] = mx; part[blockIdx.x * 2 + 1] = red[0]; }
}

// K8b: combine partials: M = max m_b; S = sum s_b * exp(m_b - M)
__global__ void k_lse_comb(const float* __restrict__ part, float* __restrict__ red_out) {
    __shared__ float rm[LSE_BLOCKS];
    __shared__ float rs[LSE_BLOCKS];
    const int t = threadIdx.x;
    const float m = part[t * 2];
    const float s = part[t * 2 + 1];
    rm[t] = m; __syncthreads();
    for (int st = LSE_BLOCKS / 2; st > 0; st >>= 1) { if (t < st) rm[t] = fmaxf(rm[t], rm[t + st]); __syncthreads(); }
    const float M = rm[0]; __syncthreads();
    rs[t] = s * expf(m - M); __syncthreads();
    for (int st = LSE_BLOCKS / 2; st > 0; st >>= 1) { if (t < st) rs[t] += rs[t + st]; __syncthreads(); }
    if (t == 0) { red_out[0] = M; red_out[1] = logf(rs[0]); }
}

// K9: out[v] = logits[v] - max - log(sum)
__global__ void k_logsoftmax(const float* __restrict__ logits, const float* __restrict__ red,
                             float* __restrict__ out) {
    const int v = blockIdx.x * 256 + threadIdx.x;
    if (v < VOCAB) out[v] = logits[v] - red[0] - red[1];
}

extern "C" void kernel_launch(void* const* d_in, const int* in_sizes, int n_in,
                              void* d_out, int out_size, void* d_ws, size_t ws_size,
                              hipStream_t stream) {
    const int*   x      = (const int*)  d_in[0];
    const float* h      = (const float*)d_in[1];
    const float* enc    = (const float*)d_in[2];
    const float* emb    = (const float*)d_in[3];
    const float* W_attn = (const float*)d_in[4];
    const float* b_attn = (const float*)d_in[5];
    const float* W_comb = (const float*)d_in[6];
    const float* b_comb = (const float*)d_in[7];
    const float* w_ih   = (const float*)d_in[8];
    const float* w_hh   = (const float*)d_in[9];
    const float* b_ih   = (const float*)d_in[10];
    const float* b_hh   = (const float*)d_in[11];
    const float* W_out  = (const float*)d_in[12];
    const float* b_out  = (const float*)d_in[13];

    float* out  = (float*)d_out;               // [VOCAB][HDIM][MAXLEN]
    float* ws   = (float*)d_ws;
    float* logits  = ws + WS_LOGITS;
    float* attnlog = ws + WS_ATTNLOG;
    float* attnw   = ws + WS_ATTNW;
    float* attnap  = ws + WS_ATTNAP;
    float* g       = ws + WS_G;
    float* gi      = ws + WS_GI;
    float* gh      = ws + WS_GH;
    float* hnew    = ws + WS_HNEW;
    float* red     = ws + WS_RED;
    float* part    = ws + WS_PART;

    k_attn_logits <<<MAXLEN, 256, 0, stream>>>(x, h, emb, W_attn, b_attn, attnlog);
    k_softmax128  <<<1, MAXLEN, 0, stream>>>(attnlog, attnw, out + VOCAB + HDIM);
    k_attn_applied<<<HDIM / 128, 128, 0, stream>>>(attnw, enc, attnap);
    k_combine     <<<HDIM / 8, 256, 0, stream>>>(x, emb, attnap, W_comb, b_comb, g);
    k_gru_matvec  <<<(6 * HDIM) / 8, 256, 0, stream>>>(g, h, w_ih, w_hh, b_ih, b_hh, gi, gh);
    k_gates       <<<(HDIM + 255) / 256, 256, 0, stream>>>(gi, gh, h, hnew, out + VOCAB);

    const int tiles  = (VOCAB + 15) / 16;      // 3142
    const int blocks = (tiles + 7) / 8;        // 393
    k_wout_wmma   <<<blocks, 256, 0, stream>>>(W_out, b_out, hnew, logits);
    k_lse_part    <<<LSE_BLOCKS, 256, 0, stream>>>(logits, part);
    k_lse_comb    <<<1, LSE_BLOCKS, 0, stream>>>(part, red);
    k_logsoftmax  <<<(VOCAB + 255) / 256, 256, 0, stream>>>(logits, red, out);
}